// task_specific_network_7172595384861
// MI455X (gfx1250) — compile-verified
//
#include <hip/hip_runtime.h>
#include <math.h>

// Problem sizes (fixed by reference).
#define BS       256
#define D_CLASS  1024
#define D_LATENT 512

typedef __attribute__((ext_vector_type(2))) float v2f;
typedef __attribute__((ext_vector_type(4))) float v4f;
typedef __attribute__((ext_vector_type(8))) float v8f;

// Numerically stable softplus matching jax.nn.softplus semantics.
__device__ __forceinline__ float softplus_f(float v) {
    return fmaxf(v, 0.0f) + log1pf(expf(-fabsf(v)));
}

// ---------------------------------------------------------------------------
// Kernel 1: elementwise prep.
//   mu[c,l]         = weight_mu[c,l] + bias_mu[c]
//   weight_var[c,l] = softplus(weight_logvar[c,l])^2 + softplus(bias_logvar[c])^2
//   sigma[c,l]      = sqrt(weight_var[c,l])           (scratch, consumed later)
// 524288 elements; float4 per thread. ~10 MB traffic -> negligible.
// ---------------------------------------------------------------------------
__global__ void prep_kernel(const float* __restrict__ wmu,
                            const float* __restrict__ wlv,
                            const float* __restrict__ bmu,
                            const float* __restrict__ blv,
                            float* __restrict__ mu_out,
                            float* __restrict__ wv_out,
                            float* __restrict__ sigma_ws) {
    int i4 = blockIdx.x * blockDim.x + threadIdx.x;      // float4 index
    int c  = i4 >> 7;                                    // (i4*4) / 512
    float bm  = bmu[c];
    float bs  = softplus_f(blv[c]);
    float bs2 = bs * bs;

    v4f wm = ((const v4f*)wmu)[i4];
    v4f wl = ((const v4f*)wlv)[i4];

    v4f mu, wv, sg;
    {
        float s;
        s = softplus_f(wl.x); wv.x = fmaf(s, s, bs2); sg.x = sqrtf(wv.x); mu.x = wm.x + bm;
        s = softplus_f(wl.y); wv.y = fmaf(s, s, bs2); sg.y = sqrtf(wv.y); mu.y = wm.y + bm;
        s = softplus_f(wl.z); wv.z = fmaf(s, s, bs2); sg.z = sqrtf(wv.z); mu.z = wm.z + bm;
        s = softplus_f(wl.w); wv.w = fmaf(s, s, bs2); sg.w = sqrtf(wv.w); mu.w = wm.w + bm;
    }
    ((v4f*)mu_out)[i4]   = mu;
    ((v4f*)wv_out)[i4]   = wv;
    ((v4f*)sigma_ws)[i4] = sg;
}

// ---------------------------------------------------------------------------
// Kernel 2: g[b,c] = sum_l x[b,l] * mu[c,l]  via V_WMMA_F32_16X16X4_F32.
// One wave per 16x16 output tile; K=512 swept in steps of 4.
// A (16x4, f32, 2 VGPRs): lane&15 = M row; lane>>4 selects K pair {0,1}/{2,3}
//   -> contiguous float2 of x[b_row, k0 + 2*half].
// B (4x16): mirror layout -> contiguous float2 of mu[c_col, k0+2*half]
//   (mu row-major == B^T, exactly what we need).
// C/D (8 VGPRs): VGPR r -> M = r + 8*half, N = lane&15.
// Writes g directly into d_out[0 : BS*D_CLASS] (kernel 3 accumulates on top).
// GEMM term is 0.27 GFLOP total -> microseconds; x and mu are L2-resident.
// ---------------------------------------------------------------------------
__global__ void gemm_mu_kernel(const float* __restrict__ x,
                               const float* __restrict__ mu,
                               float* __restrict__ out) {
    int lane = threadIdx.x;          // blockDim.x == 32 (one wave)
    int row  = lane & 15;
    int half = lane >> 4;
    int c0 = blockIdx.x * 16;
    int b0 = blockIdx.y * 16;

    const float* xrow  = x  + (size_t)(b0 + row) * D_LATENT + 2 * half;
    const float* murow = mu + (size_t)(c0 + row) * D_LATENT + 2 * half;

    v8f acc = {};
    for (int k0 = 0; k0 < D_LATENT; k0 += 4) {
        v2f a = *(const v2f*)(xrow  + k0);
        v2f b = *(const v2f*)(murow + k0);
        // 8 args: (neg_a, A, neg_b, B, c_mod, C, reuse_a, reuse_b)
        acc = __builtin_amdgcn_wmma_f32_16x16x4_f32(
            /*neg_a=*/false, a, /*neg_b=*/false, b,
            /*c_mod=*/(short)0, acc, /*reuse_a=*/false, /*reuse_b=*/false);
    }

#pragma unroll
    for (int r = 0; r < 8; ++r) {
        out[(size_t)(b0 + r + 8 * half) * D_CLASS + (c0 + row)] = acc[r];
    }
}

// ---------------------------------------------------------------------------
// Kernel 3: the bandwidth-dominant term (512 MB of eps @ 23.3 TB/s ~ 22 us).
//   out[b,c] += sum_l sigma[c,l] * eps[b,c,l] * x[b,l]
// One wave per (b,c). Lane j loads float4 at l = 4*j + 128*step -> each load
// instruction covers a contiguous 512B line of eps (perfectly coalesced b128).
// eps is single-use: loaded NON-TEMPORAL (CDNA5 TH=NT, ISA 7.3) so the 512 MB
// stream does not evict the heavily-reused sigma rows (2 MB, 256x reuse) and
// x rows (0.5 MB, 1024x reuse) from the 192 MB L2 / WGP$.
// Wave32 butterfly reduction, lane 0 accumulates onto the WMMA GEMM result.
// ---------------------------------------------------------------------------
__global__ void eps_dot_kernel(const float* __restrict__ eps,
                               const float* __restrict__ x,
                               const float* __restrict__ sigma,
                               float* __restrict__ out) {
    int gid  = blockIdx.x * blockDim.x + threadIdx.x;
    int w    = gid >> 5;                 // wave id == b*D_CLASS + c
    int lane = gid & 31;
    int b = w >> 10;                     // / D_CLASS
    int c = w & (D_CLASS - 1);

    const v4f* ep = (const v4f*)(eps   + (size_t)w * D_LATENT);
    const v4f* sg = (const v4f*)(sigma + (size_t)c * D_LATENT);
    const v4f* xr = (const v4f*)(x     + (size_t)b * D_LATENT);

    float acc = 0.0f;
#pragma unroll
    for (int j = 0; j < 4; ++j) {
        int idx = lane + j * 32;         // 128 float4s per 512-float row
        v4f e = __builtin_nontemporal_load(ep + idx);   // single-use stream
        v4f s = sg[idx];                                 // L2-resident, reused
        v4f v = xr[idx];                                 // L2-resident, reused
        acc = fmaf(e.x, s.x * v.x, acc);
        acc = fmaf(e.y, s.y * v.y, acc);
        acc = fmaf(e.z, s.z * v.z, acc);
        acc = fmaf(e.w, s.w * v.w, acc);
    }

#pragma unroll
    for (int off = 16; off > 0; off >>= 1)
        acc += __shfl_xor(acc, off, 32);

    if (lane == 0)
        out[w] += acc;
}

// ---------------------------------------------------------------------------
// Launch: inputs in setup_inputs() order:
//   0: x            [256, 512]
//   1: weight_mu    [1024, 512]
//   2: weight_logvar[1024, 512]
//   3: bias_mu      [1024, 1]
//   4: bias_logvar  [1024, 1]
//   5: eps          [256, 1024, 512]
// d_out layout (tuple, flat, return order):
//   [0            , 262144)  output     [256,1024]
//   [262144       , 786432)  mu         [1024,512]
//   [786432       , 1310720) weight_var [1024,512]
// d_ws: sigma [1024,512] = 2 MB.
// ---------------------------------------------------------------------------
extern "C" void kernel_launch(void* const* d_in, const int* in_sizes, int n_in,
                              void* d_out, int out_size, void* d_ws, size_t ws_size,
                              hipStream_t stream) {
    const float* x   = (const float*)d_in[0];
    const float* wmu = (const float*)d_in[1];
    const float* wlv = (const float*)d_in[2];
    const float* bmu = (const float*)d_in[3];
    const float* blv = (const float*)d_in[4];
    const float* eps = (const float*)d_in[5];

    float* out    = (float*)d_out;                         // [BS, D_CLASS]
    float* mu_out = out + (size_t)BS * D_CLASS;            // [D_CLASS, D_LATENT]
    float* wv_out = mu_out + (size_t)D_CLASS * D_LATENT;   // [D_CLASS, D_LATENT]
    float* sigma  = (float*)d_ws;                          // [D_CLASS, D_LATENT]

    // 1) prep: 524288 elems / 4 per thread / 256 threads = 512 blocks
    prep_kernel<<<dim3((D_CLASS * D_LATENT) / (4 * 256)), dim3(256), 0, stream>>>(
        wmu, wlv, bmu, blv, mu_out, wv_out, sigma);

    // 2) WMMA GEMM: 64 x 16 tiles of 16x16, one wave each
    gemm_mu_kernel<<<dim3(D_CLASS / 16, BS / 16), dim3(32), 0, stream>>>(
        x, mu_out, out);

    // 3) eps streaming accumulation: 262144 waves, 8 waves per block
    eps_dot_kernel<<<dim3((size_t)BS * D_CLASS * 32 / 256), dim3(256), 0, stream>>>(
        eps, x, sigma, out);
}